// Super_MambaBlock_67353677136019
// MI455X (gfx1250) — compile-verified
//
#include <hip/hip_runtime.h>
#include <hip/hip_bf16.h>
#include <math.h>

#define B_  32
#define L_  196
#define E_  384
#define D_  768
#define N_  16
#define KC_ 4          // conv kernel width
#define R_  24
#define RC_ 192
#define XD_ 56         // R + 2N

typedef __attribute__((ext_vector_type(2))) float v2f;
typedef __attribute__((ext_vector_type(8))) float v8f;

__device__ __forceinline__ float silu_f(float x) { return x / (1.0f + __expf(-x)); }
__device__ __forceinline__ float sigm_f(float x) { return 1.0f / (1.0f + __expf(-x)); }

// ---------------------------------------------------------------------------
// 16x16 f32 tile via V_WMMA_F32_16X16X4_F32 accumulated over K (K % 4 == 0).
// A: row-major 16xK (lda), rows = M.  Bs: row-major 16xK (ldb); row n of Bs
// supplies column n of the 4x16 B operand (computes A * Bs^T).
// Lane layout (ISA 7.12.2): a[v]=A[lane%16][2*(lane>>4)+v],
//                           b[v]=B[2*(lane>>4)+v][lane%16],
//                           c[v]=C[v+8*(lane>>4)][lane%16].
// ---------------------------------------------------------------------------
__device__ __forceinline__ v8f wmma_tile_full(const float* __restrict__ A, int lda,
                                              const float* __restrict__ Bs, int ldb,
                                              int K)
{
  const int lane = threadIdx.x & 31;
  const int mn   = lane & 15;
  const int g    = lane >> 4;
  const float* ar = A  + (size_t)mn * lda + 2 * g;
  const float* br = Bs + (size_t)mn * ldb + 2 * g;
  v8f acc = {0.f, 0.f, 0.f, 0.f, 0.f, 0.f, 0.f, 0.f};
#pragma unroll 8
  for (int k = 0; k < K; k += 4) {
    const v2f a = *(const v2f*)(ar + k);     // 8B aligned: lda,2g,k all even
    const v2f b = *(const v2f*)(br + k);
    acc = __builtin_amdgcn_wmma_f32_16x16x4_f32(false, a, false, b, (short)0, acc,
                                                false, false);
  }
  return acc;
}

// Edge-tile variant: rows of Bs at/after nvalid contribute zero. The address is
// clamped in-bounds and the value zeroed with a branchless select (v_cndmask),
// so the inner loop stays unconditional vector loads + wmma.
__device__ __forceinline__ v8f wmma_tile_masked(const float* __restrict__ A, int lda,
                                                const float* __restrict__ Bs, int ldb,
                                                int nvalid, int K)
{
  const int lane = threadIdx.x & 31;
  const int mn   = lane & 15;
  const int g    = lane >> 4;
  const bool ok  = mn < nvalid;
  const int  bn  = ok ? mn : (nvalid - 1);   // clamped, always in-bounds
  const float* ar = A  + (size_t)mn * lda + 2 * g;
  const float* br = Bs + (size_t)bn * ldb + 2 * g;
  v8f acc = {0.f, 0.f, 0.f, 0.f, 0.f, 0.f, 0.f, 0.f};
#pragma unroll 8
  for (int k = 0; k < K; k += 4) {
    const v2f a = *(const v2f*)(ar + k);
    v2f b = *(const v2f*)(br + k);
    b.x = ok ? b.x : 0.f;
    b.y = ok ? b.y : 0.f;
    acc = __builtin_amdgcn_wmma_f32_16x16x4_f32(false, a, false, b, (short)0, acc,
                                                false, false);
  }
  return acc;
}

// ---------------- in_proj: xz[b,d,l] = sum_e W[d,e]*hs[b,l,e] -----------------
__global__ void k_in_proj(const float* __restrict__ hs, const float* __restrict__ W,
                          float* __restrict__ xz)
{
  const int wave = threadIdx.x >> 5;
  const int tile = blockIdx.x * 8 + wave;      // 96*13 = 1248 tiles per batch
  const int b  = blockIdx.y;
  const int tm = tile / 13;
  const int tn = tile % 13;
  const int n0 = tn * 16;
  const int nvalid = (L_ - n0 < 16) ? (L_ - n0) : 16;
  const float* A  = W  + (size_t)(tm * 16) * E_;
  const float* Bs = hs + (size_t)b * L_ * E_ + (size_t)n0 * E_;
  const v8f acc = (nvalid == 16) ? wmma_tile_full(A, E_, Bs, E_, E_)
                                 : wmma_tile_masked(A, E_, Bs, E_, nvalid, E_);
  const int lane = threadIdx.x & 31;
  const int n = lane & 15, g = lane >> 4;
  if (n < nvalid) {
    float* C = xz + (size_t)b * (2 * D_) * L_ + (size_t)(tm * 16) * L_ + n0 + n;
#pragma unroll
    for (int v = 0; v < 8; ++v) C[(size_t)(v + 8 * g) * L_] = acc[v];
  }
}

// ------------- depthwise causal conv + SiLU (with optional time flip) ---------
__global__ void k_conv(const float* __restrict__ xz, const float* __restrict__ cw,
                       const float* __restrict__ cb, float* __restrict__ u, int dir)
{
  const int i = blockIdx.x * blockDim.x + threadIdx.x;   // over B*L*D
  const int d = i % D_;
  const int t = (i / D_) % L_;
  const int b = i / (D_ * L_);
  const float* x = xz + (size_t)b * (2 * D_) * L_ + (size_t)d * L_;
  float acc = cb[dir * D_ + d];
#pragma unroll
  for (int j = 0; j < KC_; ++j) {
    const int tp = t - (KC_ - 1) + j;
    if (tp >= 0) {
      const int l = dir ? (L_ - 1 - tp) : tp;
      acc += cw[(dir * D_ + d) * KC_ + j] * x[l];
    }
  }
  u[i] = silu_f(acc);
}

// ----------- xproj: xdbl[m,r] = sum_d u[m,d] * xproj_w[dir,r,d] ---------------
__global__ void k_xproj(const float* __restrict__ u, const float* __restrict__ W,
                        float* __restrict__ xdbl, int dir)
{
  const int wave = threadIdx.x >> 5;
  const int tile = blockIdx.x * 8 + wave;      // 392*4 tiles
  const int tm = tile >> 2;
  const int tn = tile & 3;
  const int n0 = tn * 16;
  const int nvalid = (XD_ - n0 < 16) ? (XD_ - n0) : 16;
  const float* A  = u + (size_t)(tm * 16) * D_;
  const float* Bs = W + (size_t)dir * XD_ * D_ + (size_t)n0 * D_;
  const v8f acc = (nvalid == 16) ? wmma_tile_full(A, D_, Bs, D_, D_)
                                 : wmma_tile_masked(A, D_, Bs, D_, nvalid, D_);
  const int lane = threadIdx.x & 31;
  const int n = lane & 15, g = lane >> 4;
  if (n < nvalid) {
    float* C = xdbl + (size_t)(tm * 16) * XD_ + n0 + n;
#pragma unroll
    for (int v = 0; v < 8; ++v) C[(size_t)(v + 8 * g) * XD_] = acc[v];
  }
}

// ----- dtproj + softplus: dt[m,d] = softplus(sum_r xdbl[m,r]*W[dir,d,r]+b) ----
__global__ void k_dtproj(const float* __restrict__ xdbl, const float* __restrict__ W,
                         const float* __restrict__ bias, float* __restrict__ dt, int dir)
{
  const int wave = threadIdx.x >> 5;
  const int tile = blockIdx.x * 8 + wave;      // 392*48 tiles
  const int tm = tile / 48;
  const int tn = tile % 48;
  const float* A  = xdbl + (size_t)(tm * 16) * XD_;
  const float* Bs = W + (size_t)dir * D_ * R_ + (size_t)(tn * 16) * R_;
  const v8f acc = wmma_tile_full(A, XD_, Bs, R_, R_);
  const int lane = threadIdx.x & 31;
  const int n = lane & 15, g = lane >> 4;
  const int d = tn * 16 + n;
  const float bv = bias[dir * D_ + d];
  float* C = dt + (size_t)(tm * 16) * D_ + d;
#pragma unroll
  for (int v = 0; v < 8; ++v) {
    const float x = acc[v] + bv;
    C[(size_t)(v + 8 * g) * D_] = (x > 20.f) ? x : log1pf(__expf(x));
  }
}

// --------------------------- selective scan + gate ----------------------------
// grid (D/16, B), block 256 (8 waves); lane = n + 16*(d parity); 2 channels/wave
__global__ void k_scan(const float* __restrict__ u, const float* __restrict__ dt,
                       const float* __restrict__ xdbl, const float* __restrict__ A_log,
                       const float* __restrict__ Dp, const float* __restrict__ xz,
                       float* __restrict__ o, int dir)
{
  const int lane = threadIdx.x & 31;
  const int wave = threadIdx.x >> 5;
  const int n = lane & 15;
  const int dsub = (lane >> 4) + wave * 2;     // 0..15
  const int d = blockIdx.x * 16 + dsub;
  const int b = blockIdx.y;
  const float Aval = -__expf(A_log[(dir * D_ + d) * N_ + n]);
  const float Dv = Dp[dir * D_ + d];
  const float* zrow  = xz + (size_t)b * (2 * D_) * L_ + (size_t)(D_ + d) * L_;
  const float* urow  = u  + (size_t)b * L_ * D_ + d;
  const float* dtrow = dt + (size_t)b * L_ * D_ + d;
  const float* xrow  = xdbl + (size_t)b * L_ * XD_;
  float h = 0.f;
  for (int t = 0; t < L_; ++t) {
    const float dtv = dtrow[(size_t)t * D_];
    const float uv  = urow[(size_t)t * D_];
    const float Bv  = xrow[t * XD_ + R_ + n];
    const float Cv  = xrow[t * XD_ + R_ + N_ + n];
    h = __expf(dtv * Aval) * h + (dtv * uv) * Bv;
    float yp = h * Cv;
    yp += __shfl_xor(yp, 1);
    yp += __shfl_xor(yp, 2);
    yp += __shfl_xor(yp, 4);
    yp += __shfl_xor(yp, 8);
    if (n == 0) {
      const int l = dir ? (L_ - 1 - t) : t;
      const float z = zrow[l];
      o[((size_t)b * L_ + l) * D_ + d] = (yp + uv * Dv) * silu_f(z);
    }
  }
}

// ------------- BiAttn part 1: layernorm + mean over L into g[b,d] -------------
__global__ void k_lnmean(const float* __restrict__ o, const float* __restrict__ ls,
                         const float* __restrict__ lb, float* __restrict__ g, int dir)
{
  __shared__ float s1[256], s2[256];
  const int b = blockIdx.y, l = blockIdx.x, tid = threadIdx.x;
  const float* x = o + ((size_t)b * L_ + l) * D_;
  const float v0 = x[tid], v1 = x[tid + 256], v2 = x[tid + 512];
  s1[tid] = v0 + v1 + v2;
  s2[tid] = v0 * v0 + v1 * v1 + v2 * v2;
  __syncthreads();
  for (int st = 128; st > 0; st >>= 1) {
    if (tid < st) { s1[tid] += s1[tid + st]; s2[tid] += s2[tid + st]; }
    __syncthreads();
  }
  const float mu  = s1[0] * (1.f / D_);
  const float var = s2[0] * (1.f / D_) - mu * mu;
  const float rs  = rsqrtf(var + 1e-5f);
  const float invL = 1.f / L_;
#pragma unroll
  for (int j = 0; j < 3; ++j) {
    const int d = tid + j * 256;
    const float vv = (j == 0) ? v0 : ((j == 1) ? v1 : v2);
    const float xn = (vv - mu) * rs * ls[dir * D_ + d] + lb[dir * D_ + d];
    atomicAdd(&g[b * D_ + d], xn * invL);
  }
}

// ---------- BiAttn part 2: attn[b,d] = sigmoid(cs(gelu(gr(g)))) ---------------
__global__ void k_attn(const float* __restrict__ g, const float* __restrict__ grw,
                       const float* __restrict__ grb, const float* __restrict__ csw,
                       const float* __restrict__ csb, float* __restrict__ attn, int dir)
{
  __shared__ float gs[RC_];
  const int b = blockIdx.x, tid = threadIdx.x;
  const float* gb = g + (size_t)b * D_;
  if (tid < RC_) {
    const float* wr = grw + ((size_t)dir * RC_ + tid) * D_;
    float acc = grb[dir * RC_ + tid];
    for (int k = 0; k < D_; ++k) acc += gb[k] * wr[k];
    const float t = tanhf(0.7978845608028654f * (acc + 0.044715f * acc * acc * acc));
    gs[tid] = 0.5f * acc * (1.f + t);     // gelu (tanh approx, jax default)
  }
  __syncthreads();
  for (int d = tid; d < D_; d += 256) {
    const float* wc = csw + ((size_t)dir * D_ + d) * RC_;
    float acc = csb[dir * D_ + d];
    for (int k = 0; k < RC_; ++k) acc += gs[k] * wc[k];
    attn[b * D_ + d] = sigm_f(acc);
  }
}

// ------------- apply channel attention, accumulate across directions ----------
__global__ void k_apply(const float* __restrict__ o, const float* __restrict__ attn,
                        float* __restrict__ osum, int accum)
{
  const int i = blockIdx.x * blockDim.x + threadIdx.x;   // over B*L*D
  const int d = i % D_;
  const int b = i / (L_ * D_);
  const float v = o[i] * attn[b * D_ + d];
  osum[i] = accum ? (osum[i] + v) : v;
}

// -------- out_proj: out[m,e] = sum_d osum[m,d] * out_proj_w[e,d] --------------
__global__ void k_outproj(const float* __restrict__ osum, const float* __restrict__ W,
                          float* __restrict__ out)
{
  const int wave = threadIdx.x >> 5;
  const int tile = blockIdx.x * 8 + wave;      // 392*24 tiles
  const int tm = tile / 24;
  const int tn = tile % 24;
  const float* A  = osum + (size_t)(tm * 16) * D_;
  const float* Bs = W + (size_t)(tn * 16) * D_;
  const v8f acc = wmma_tile_full(A, D_, Bs, D_, D_);
  const int lane = threadIdx.x & 31;
  const int n = lane & 15, g = lane >> 4;
  float* C = out + (size_t)(tm * 16) * E_ + tn * 16 + n;
#pragma unroll
  for (int v = 0; v < 8; ++v) C[(size_t)(v + 8 * g) * E_] = acc[v];
}

extern "C" void kernel_launch(void* const* d_in, const int* in_sizes, int n_in,
                              void* d_out, int out_size, void* d_ws, size_t ws_size,
                              hipStream_t stream) {
  (void)in_sizes; (void)n_in; (void)out_size; (void)ws_size;
  const float* hs   = (const float*)d_in[0];
  const float* Win  = (const float*)d_in[1];
  const float* Wout = (const float*)d_in[2];
  const float* Alog = (const float*)d_in[3];
  const float* cw   = (const float*)d_in[4];
  const float* cb   = (const float*)d_in[5];
  const float* xpw  = (const float*)d_in[6];
  const float* dtw  = (const float*)d_in[7];
  const float* dtb  = (const float*)d_in[8];
  const float* Dpar = (const float*)d_in[9];
  const float* lns  = (const float*)d_in[10];
  const float* lnb  = (const float*)d_in[11];
  const float* grw  = (const float*)d_in[12];
  const float* grb  = (const float*)d_in[13];
  const float* csw  = (const float*)d_in[14];
  const float* csb  = (const float*)d_in[15];
  float* out = (float*)d_out;

  // workspace layout (floats)
  float* ws = (float*)d_ws;
  const size_t SZ_XZ  = (size_t)B_ * 2 * D_ * L_;   // 9,633,792
  const size_t SZ_BLD = (size_t)B_ * L_ * D_;       // 4,816,896
  const size_t SZ_XDB = (size_t)B_ * L_ * XD_;      //   351,232
  float* xz   = ws;
  float* u    = xz + SZ_XZ;
  float* xdbl = u + SZ_BLD;
  float* dt   = xdbl + SZ_XDB;
  float* o    = dt + SZ_BLD;
  float* g    = o + SZ_BLD;
  float* attn = g + (size_t)B_ * D_;
  float* osum = attn + (size_t)B_ * D_;

  const int elemsBLD = B_ * L_ * D_;

  k_in_proj<<<dim3(156, 32), 256, 0, stream>>>(hs, Win, xz);

  for (int dir = 0; dir < 2; ++dir) {
    k_conv<<<elemsBLD / 256, 256, 0, stream>>>(xz, cw, cb, u, dir);
    k_xproj<<<196, 256, 0, stream>>>(u, xpw, xdbl, dir);
    k_dtproj<<<2352, 256, 0, stream>>>(xdbl, dtw, dtb, dt, dir);
    k_scan<<<dim3(48, 32), 256, 0, stream>>>(u, dt, xdbl, Alog, Dpar, xz, o, dir);
    hipMemsetAsync(g, 0, (size_t)B_ * D_ * sizeof(float), stream);
    k_lnmean<<<dim3(196, 32), 256, 0, stream>>>(o, lns, lnb, g, dir);
    k_attn<<<32, 256, 0, stream>>>(g, grw, grb, csw, csb, attn, dir);
    k_apply<<<elemsBLD / 256, 256, 0, stream>>>(o, attn, osum, dir);
  }

  k_outproj<<<1176, 256, 0, stream>>>(osum, Wout, out);
}